// UnitaryLayer_82102594830860
// MI455X (gfx1250) — compile-verified
//
#include <hip/hip_runtime.h>
#include <math.h>

typedef float v2f __attribute__((ext_vector_type(2)));
typedef float v8f __attribute__((ext_vector_type(8)));
typedef unsigned int v4u __attribute__((ext_vector_type(4)));
typedef int v4i __attribute__((ext_vector_type(4)));
typedef int v8i __attribute__((ext_vector_type(8)));

#define STEP_ANG 0.024543692606170259f  /* 2*pi/256 */

static __device__ __forceinline__ v8f wmma4(v2f a, v2f b, v8f c) {
    // V_WMMA_F32_16X16X4_F32 : D = A(16x4) * B(4x16) + C(16x16)
    return __builtin_amdgcn_wmma_f32_16x16x4_f32(
        false, a, false, b, (short)0, c, false, false);
}

/* TDM 2D tile load: 16 rows x 128 fp32, row stride 65536 elements,
   LDS padded +4 DWORDs every 128 DWORDs (-> LDS row stride 132). */
static __device__ __forceinline__ void tdm_load_x_tile(unsigned int lds_off,
                                                       unsigned long long gaddr) {
    v4u g0;
    g0[0] = 1u;                                   // count=1 valid descriptor
    g0[1] = lds_off;                              // lds_addr (bytes)
    g0[2] = (unsigned int)(gaddr & 0xffffffffull);
    g0[3] = (unsigned int)((gaddr >> 32) & 0x01ffffffull) | (2u << 30); // type=2
    v8i g1;
    g1[0] = (int)((2u << 16)      // data_size = 4B
                | (1u << 20)      // pad_enable
                | (6u << 22)      // pad_interval: 128 DWORDs
                | (3u << 25));    // pad_amount : 4 DWORDs
    g1[1] = (int)(128u << 16);    // tensor_dim0 = 128 (bits 79:48 low half)
    g1[2] = 16;                   // tensor_dim1 = 16
    g1[3] = (int)(128u << 16);    // tile_dim0 = 128
    g1[4] = 16;                   // tile_dim1 = 16, tile_dim2 = 0
    g1[5] = 65536;                // tensor_dim0_stride = 65536 elements
    g1[6] = 0;
    g1[7] = 0;
    v4i z4 = {0, 0, 0, 0};
#if defined(__clang_major__) && __clang_major__ >= 23
    v8i z8 = {0, 0, 0, 0, 0, 0, 0, 0};
    __builtin_amdgcn_tensor_load_to_lds(g0, g1, z4, z4, z8, 0);
#else
    __builtin_amdgcn_tensor_load_to_lds(g0, g1, z4, z4, 0);
#endif
}

/* ---------------- W composition (tiny, one-time) ----------------
   A'[j,k] = exp(i*(d1_j - 2pi*j*perm[k]/256))        (fft(D1) + col-perm)
   t2 = R1 @ A' ; u = D2 t2 ; t3 = u @ Finv (row iDFT) ; W = D3 (R2 @ t3)  */

__global__ void compose_c1(const float* __restrict__ d1,
                           const float* __restrict__ r1_re,
                           const float* __restrict__ r1_im,
                           const int*   __restrict__ perm,
                           float* __restrict__ wv) {
    int k = threadIdx.x;
    int pk = perm[k];
    float wr = 0.f, wi = 0.f, n1 = 0.f;
    for (int j = 0; j < 256; ++j) {
        float th = d1[j] - STEP_ANG * (float)((j * pk) & 255);
        float c, s; sincosf(th, &s, &c);
        float vr = r1_re[j], vi = r1_im[j];
        wr += vr * c + vi * s;          // conj(v1)_j * A'[j,k]
        wi += vr * s - vi * c;
        n1 += vr * vr + vi * vi;
    }
    wv[k] = wr; wv[256 + k] = wi;
    if (k == 0) wv[1024] = n1;
}

__global__ void compose_c2(const float* __restrict__ d1,
                           const float* __restrict__ d2,
                           const float* __restrict__ r1_re,
                           const float* __restrict__ r1_im,
                           const int*   __restrict__ perm,
                           const float* __restrict__ wv,
                           float* __restrict__ u_re,
                           float* __restrict__ u_im) {
    int j = blockIdx.x, k = threadIdx.x;
    int pk = perm[k];
    float th = d1[j] - STEP_ANG * (float)((j * pk) & 255);
    float c, s; sincosf(th, &s, &c);
    float sc = 2.0f / wv[1024];
    float vr = r1_re[j], vi = r1_im[j];
    float wr = wv[k], wi = wv[256 + k];
    float t2r = c - sc * (vr * wr - vi * wi);
    float t2i = s - sc * (vr * wi + vi * wr);
    float c2, s2; sincosf(d2[j], &s2, &c2);
    u_re[j * 256 + k] = c2 * t2r - s2 * t2i;
    u_im[j * 256 + k] = c2 * t2i + s2 * t2r;
}

__global__ void compose_c3(const float* __restrict__ u_re,
                           const float* __restrict__ u_im,
                           float* __restrict__ t3_re,
                           float* __restrict__ t3_im) {
    int j = blockIdx.x, k = threadIdx.x;
    float tr = 0.f, ti = 0.f;
    for (int m = 0; m < 256; ++m) {       // row-wise inverse DFT
        float ang = STEP_ANG * (float)((m * k) & 255);
        float c, s; sincosf(ang, &s, &c);
        float ur = u_re[j * 256 + m], ui = u_im[j * 256 + m];
        tr += ur * c - ui * s;
        ti += ur * s + ui * c;
    }
    t3_re[j * 256 + k] = tr * (1.0f / 256.0f);
    t3_im[j * 256 + k] = ti * (1.0f / 256.0f);
}

__global__ void compose_c4(const float* __restrict__ t3_re,
                           const float* __restrict__ t3_im,
                           const float* __restrict__ r2_re,
                           const float* __restrict__ r2_im,
                           float* __restrict__ wv) {
    int k = threadIdx.x;
    float wr = 0.f, wi = 0.f, n2 = 0.f;
    for (int j = 0; j < 256; ++j) {
        float vr = r2_re[j], vi = r2_im[j];
        float tr = t3_re[j * 256 + k], ti = t3_im[j * 256 + k];
        wr += vr * tr + vi * ti;
        wi += vr * ti - vi * tr;
        n2 += vr * vr + vi * vi;
    }
    wv[512 + k] = wr; wv[768 + k] = wi;
    if (k == 0) wv[1025] = n2;
}

/* Final reflector+phase; writes W in WMMA-B swizzled layout:
   Wsw[((kdim>>1)*256 + n)*2 + (kdim&1)], plus the Gauss sum plane. */
__global__ void compose_c5(const float* __restrict__ t3_re,
                           const float* __restrict__ t3_im,
                           const float* __restrict__ d3,
                           const float* __restrict__ r2_re,
                           const float* __restrict__ r2_im,
                           const float* __restrict__ wv,
                           float* __restrict__ Wre_sw,
                           float* __restrict__ Wim_sw,
                           float* __restrict__ Wsum_sw) {
    int j = blockIdx.x, k = threadIdx.x;   // j = GEMM k-dim, k = n-dim
    float sc = 2.0f / wv[1025];
    float vr = r2_re[j], vi = r2_im[j];
    float wr = wv[512 + k], wi = wv[768 + k];
    float tr = t3_re[j * 256 + k] - sc * (vr * wr - vi * wi);
    float ti = t3_im[j * 256 + k] - sc * (vr * wi + vi * wr);
    float c, s; sincosf(d3[j], &s, &c);
    float ore = c * tr - s * ti;
    float oim = c * ti + s * tr;
    int sw = (((j >> 1) << 8) + k) * 2 + (j & 1);
    Wre_sw[sw]  = ore;
    Wim_sw[sw]  = oim;
    Wsum_sw[sw] = ore + oim;
}

/* Swizzle V (I=128 x U=256) into WMMA-B layout for the h-GEMM. */
__global__ void vswizzle(const float* __restrict__ V_re,
                         const float* __restrict__ V_im,
                         float* __restrict__ Vre_sw,
                         float* __restrict__ Vim_sw) {
    int i = blockIdx.x, u = threadIdx.x;
    int sw = (((i >> 1) << 8) + u) * 2 + (i & 1);
    Vre_sw[sw] = V_re[i * 256 + u];
    Vim_sw[sw] = V_im[i * 256 + u];
}

/* ------------- h_all = x @ V (complex fp32 WMMA GEMM) -------------
   grid (4, 512): blockIdx.x = 16-row batch tile, blockIdx.y = t.
   x tiles staged via Tensor Data Mover with LDS padding.           */
__global__ void hgemm_kernel(const float* __restrict__ x_re,
                             const float* __restrict__ x_im,
                             const float* __restrict__ Vre_sw,
                             const float* __restrict__ Vim_sw,
                             float2* __restrict__ out) {
    const int t  = blockIdx.y;
    const int b0 = blockIdx.x * 16;
    __shared__ float xs_re[16][132];   // 132 = TDM pad(128+4): conflict-free cols
    __shared__ float xs_im[16][132];
    const int tid = threadIdx.x;
    if (tid < 32) {                    // one wave drives the TDM
        unsigned long long off = ((unsigned long long)(b0 * 512 + t) * 128ull) * 4ull;
        tdm_load_x_tile((unsigned int)(uintptr_t)(&xs_re[0][0]),
                        (unsigned long long)(uintptr_t)x_re + off);
        tdm_load_x_tile((unsigned int)(uintptr_t)(&xs_im[0][0]),
                        (unsigned long long)(uintptr_t)x_im + off);
        __builtin_amdgcn_s_wait_tensorcnt(0);
    }
    __syncthreads();
    const int lane = tid & 31;
    const int half = lane >> 4;
    const int ln16 = lane & 15;
    const int n0   = (tid >> 5) * 32;
    const v2f* Vre2 = (const v2f*)Vre_sw;
    const v2f* Vim2 = (const v2f*)Vim_sw;
    v8f rr0 = {}, ii0 = {}, ri0 = {}, ir0 = {};
    v8f rr1 = {}, ii1 = {}, ri1 = {}, ir1 = {};
    for (int kb = 0; kb < 32; ++kb) {
        int k0 = 4 * kb + 2 * half;
        v2f are, aim;
        are.x = xs_re[ln16][k0]; are.y = xs_re[ln16][k0 + 1];
        aim.x = xs_im[ln16][k0]; aim.y = xs_im[ln16][k0 + 1];
        int bi = (2 * kb + half) * 256 + n0 + ln16;
        v2f bre0 = Vre2[bi];      v2f bim0 = Vim2[bi];
        v2f bre1 = Vre2[bi + 16]; v2f bim1 = Vim2[bi + 16];
        rr0 = wmma4(are, bre0, rr0);  ii0 = wmma4(aim, bim0, ii0);
        ri0 = wmma4(are, bim0, ri0);  ir0 = wmma4(aim, bre0, ir0);
        rr1 = wmma4(are, bre1, rr1);  ii1 = wmma4(aim, bim1, ii1);
        ri1 = wmma4(are, bim1, ri1);  ir1 = wmma4(aim, bre1, ir1);
    }
    #pragma unroll
    for (int r = 0; r < 8; ++r) {
        int m = r + 8 * half;
        int base = (t * 64 + (b0 + m)) * 256;
        out[base + n0 + ln16]      = make_float2(rr0[r] - ii0[r], ri0[r] + ir0[r]);
        out[base + n0 + 16 + ln16] = make_float2(rr1[r] - ii1[r], ri1[r] + ir1[r]);
    }
}

/* -------- sequential scan: state = modReLU(h_t + state @ W) --------
   4 persistent workgroups (16 batch rows each), 512 thr = 16 waves,
   wave w owns output columns [16w,16w+16). Gauss 3-mult complex GEMM:
   re = rr - ii ; im = ss - rr - ii. State in LDS; W resident in L2.  */
__global__ void scan_kernel(const float* __restrict__ Wre_sw,
                            const float* __restrict__ Wim_sw,
                            const float* __restrict__ Wsum_sw,
                            const float* __restrict__ bias,
                            const float* __restrict__ x0_re,
                            const float* __restrict__ x0_im,
                            float2* __restrict__ out) {
    const int b0 = blockIdx.x * 16;
    __shared__ float st_re[16][260];
    __shared__ float st_im[16][260];
    __shared__ float st_sum[16][260];
    const int tid = threadIdx.x;
    for (int off = tid * 32; off < 65536; off += 512 * 32) {  // warm L2 with W
        __builtin_prefetch(Wre_sw + off, 0, 3);
        __builtin_prefetch(Wim_sw + off, 0, 3);
        __builtin_prefetch(Wsum_sw + off, 0, 3);
    }
    #pragma unroll
    for (int r = 0; r < 8; ++r) {
        int e = tid + 512 * r;         // state0 = broadcast(x0)
        int m = e >> 8, u = e & 255;
        float a = x0_re[u], b = x0_im[u];
        st_re[m][u] = a;
        st_im[m][u] = b;
        st_sum[m][u] = a + b;
    }
    __syncthreads();
    const int lane = tid & 31;
    const int half = lane >> 4;
    const int ln16 = lane & 15;
    const int u    = (tid >> 5) * 16 + ln16;
    const float bcol = bias[u];
    const v2f* Wre2  = (const v2f*)Wre_sw;
    const v2f* Wim2  = (const v2f*)Wim_sw;
    const v2f* Wsum2 = (const v2f*)Wsum_sw;
    for (int t = 0; t < 512; ++t) {
        v8f rr = {}, ii = {}, ss = {};
        for (int kb = 0; kb < 64; ++kb) {
            int k0 = 4 * kb + 2 * half;
            v2f are, aim, asm_;
            are.x  = st_re[ln16][k0];  are.y  = st_re[ln16][k0 + 1];
            aim.x  = st_im[ln16][k0];  aim.y  = st_im[ln16][k0 + 1];
            asm_.x = st_sum[ln16][k0]; asm_.y = st_sum[ln16][k0 + 1];
            int bi = (2 * kb + half) * 256 + u;
            rr = wmma4(are,  Wre2[bi],  rr);
            ii = wmma4(aim,  Wim2[bi],  ii);
            ss = wmma4(asm_, Wsum2[bi], ss);
        }
        __syncthreads();               // all state reads done
        #pragma unroll
        for (int r = 0; r < 8; ++r) {
            int m = r + 8 * half;
            int idx = (t * 64 + (b0 + m)) * 256 + u;
            float2 h = out[idx];       // h_all staged in d_out, consumed once
            float ore = rr[r] - ii[r] + h.x;
            float oim = ss[r] - rr[r] - ii[r] + h.y;
            float nrm = sqrtf(ore * ore + oim * oim);
            float Mv  = fmaxf(nrm + bcol, 0.0f) / nrm;   // modReLU scale
            float nre = Mv * ore, nim = Mv * oim;
            out[idx] = make_float2(nre, nim);
            st_re[m][u]  = nre;
            st_im[m][u]  = nim;
            st_sum[m][u] = nre + nim;
        }
        __syncthreads();               // state fully updated for next step
    }
}

extern "C" void kernel_launch(void* const* d_in, const int* in_sizes, int n_in,
                              void* d_out, int out_size, void* d_ws, size_t ws_size,
                              hipStream_t stream) {
    const float* x_re  = (const float*)d_in[0];
    const float* x_im  = (const float*)d_in[1];
    const float* d1    = (const float*)d_in[2];
    const float* d2    = (const float*)d_in[3];
    const float* d3    = (const float*)d_in[4];
    const float* r1_re = (const float*)d_in[5];
    const float* r1_im = (const float*)d_in[6];
    const float* r2_re = (const float*)d_in[7];
    const float* r2_im = (const float*)d_in[8];
    const float* V_re  = (const float*)d_in[9];
    const float* V_im  = (const float*)d_in[10];
    const float* bias  = (const float*)d_in[11];
    const float* x0_re = (const float*)d_in[12];
    const float* x0_im = (const float*)d_in[13];
    const int*   perm  = (const int*)d_in[14];

    float* ws      = (float*)d_ws;     // ~2.1 MB used
    float* u_re    = ws;
    float* u_im    = ws + 65536;
    float* t3_re   = ws + 131072;
    float* t3_im   = ws + 196608;
    float* Wre_sw  = ws + 262144;
    float* Wim_sw  = ws + 327680;
    float* Wsum_sw = ws + 393216;
    float* Vre_sw  = ws + 458752;
    float* Vim_sw  = ws + 491520;
    float* wv      = ws + 524288;      // w1(512) w2(512) n1 n2
    float2* out    = (float2*)d_out;

    compose_c1<<<1, 256, 0, stream>>>(d1, r1_re, r1_im, perm, wv);
    compose_c2<<<256, 256, 0, stream>>>(d1, d2, r1_re, r1_im, perm, wv, u_re, u_im);
    compose_c3<<<256, 256, 0, stream>>>(u_re, u_im, t3_re, t3_im);
    compose_c4<<<1, 256, 0, stream>>>(t3_re, t3_im, r2_re, r2_im, wv);
    compose_c5<<<256, 256, 0, stream>>>(t3_re, t3_im, d3, r2_re, r2_im, wv,
                                        Wre_sw, Wim_sw, Wsum_sw);
    vswizzle<<<128, 256, 0, stream>>>(V_re, V_im, Vre_sw, Vim_sw);
    hgemm_kernel<<<dim3(4, 512), 256, 0, stream>>>(x_re, x_im, Vre_sw, Vim_sw, out);
    scan_kernel<<<4, 512, 0, stream>>>(Wre_sw, Wim_sw, Wsum_sw, bias,
                                       x0_re, x0_im, out);
}